// NestedGraphTitanV6_34402688040970
// MI455X (gfx1250) — compile-verified
//
#include <hip/hip_runtime.h>
#include <math.h>

// ---------------------------------------------------------------------------
// NestedGraphTitanV6 forward for MI455X (gfx1250, wave32).
// - GEMMs: V_WMMA_F32_16X16X4_F32 (exact fp32 on the matrix pipe)
// - K=128 GEMMs: block's 16x128 A-tile DMA'd to LDS by the Tensor Data Mover
//   (tensor_load_to_lds + s_wait_tensorcnt), padded rows (130 dwords) for
//   conflict-free half-wave ds_load_b64 reads. Compile-time specialization
//   keeps the K-loop branch-free and fully unrolled.
// - Sequential delta-rule scan keeps its 128x128 state matrix in VGPRs,
//   wave32 shuffle reductions, global_prefetch for the next timestep.
// ---------------------------------------------------------------------------

typedef __attribute__((ext_vector_type(2))) float v2f;
typedef __attribute__((ext_vector_type(8))) float v8f;
typedef __attribute__((ext_vector_type(4))) unsigned int v4u;
typedef __attribute__((ext_vector_type(8))) int v8i;
typedef __attribute__((ext_vector_type(4))) int v4i;

#define DD   128
#define ROWS 16384   // B*S*N = 64*64*4
#define EPS  1e-5f

__device__ __forceinline__ float sigmoidf_(float x) { return 1.0f / (1.0f + __expf(-x)); }
__device__ __forceinline__ float siluf_(float x)    { return x * sigmoidf_(x); }
__device__ __forceinline__ float geluf_(float x)    { return 0.5f * x * (1.0f + erff(x * 0.7071067811865475f)); }

// All 128 threads of the block must call.
__device__ __forceinline__ float block_reduce128(float v, float* red, int t) {
    red[t] = v; __syncthreads();
    #pragma unroll
    for (int s = 64; s > 0; s >>= 1) {
        if (t < s) red[t] += red[t + s];
        __syncthreads();
    }
    float r = red[0];
    __syncthreads();
    return r;
}

// ---------------------------------------------------------------------------
// Generic fp32 WMMA GEMM: C[M,N] = act(A[M,K] @ B[K,N] + bias)
// one wave -> one 16x16 tile; K stepped by 4 through v_wmma_f32_16x16x4_f32.
// TDM=true: K is compile-time 128 and the block's A-tile is fetched to LDS
// once by the Tensor Data Mover. TDM=false: runtime K, per-lane global loads.
// act: 0 = none, 1 = silu
// ---------------------------------------------------------------------------
template <bool TDM>
__global__ __launch_bounds__(256) void gemm_wmma_f32(
    const float* __restrict__ A, const float* __restrict__ B,
    const float* __restrict__ bias, float* __restrict__ C,
    int M, int N, int K, int act)
{
    __shared__ float sA[16 * 130];            // 16 rows x (128 + 2 pad) dwords
    const int wpb  = blockDim.x >> 5;
    const int wave = threadIdx.x >> 5;
    const int lane = threadIdx.x & 31;
    const int mt   = blockIdx.x;
    const int nt   = blockIdx.y * wpb + wave;
    const int row0 = mt * 16;
    const int m  = lane & 15;
    const int ko = (lane >> 4) << 1;          // 0 or 2 (K sub-slot per half-wave)

    if (TDM) {
        if (threadIdx.x < 32) {               // wave 0 issues the tensor DMA
            unsigned long long ga = (unsigned long long)(size_t)(A + (size_t)row0 * 128);
            unsigned ldsa = (unsigned)(size_t)(void*)sA;
            v4u g0; v8i g1; v4i g2, g3;
            g0[0] = 1u;                                            // count=1, user mode
            g0[1] = ldsa;                                          // lds_addr
            g0[2] = (unsigned)(ga & 0xFFFFFFFFu);                  // global_addr[31:0]
            g0[3] = (unsigned)((ga >> 32) & 0x01FFFFFFu) | (2u << 30);  // addr[56:32] | type=2
            // data_size=4B(2), pad_enable, pad_interval=128dw(6), pad_amount=2dw(1)
            g1[0] = (int)((2u << 16) | (1u << 20) | (6u << 22) | (1u << 25));
            g1[1] = (int)(128u << 16);                             // tensor_dim0 = 128
            g1[2] = (int)(16u  << 16);                             // tensor_dim1 = 16
            g1[3] = (int)(128u << 16);                             // tile_dim0 = 128
            g1[4] = (int)16;                                       // tile_dim1 = 16, tile_dim2 = 0
            g1[5] = (int)128;                                      // tensor_dim0_stride = 128
            g1[6] = 0; g1[7] = 0;
            g2[0] = 0; g2[1] = 0; g2[2] = 0; g2[3] = 0;
            g3 = g2;
#if defined(__clang_major__) && (__clang_major__ >= 23)
            v8i gz; gz[0]=0; gz[1]=0; gz[2]=0; gz[3]=0; gz[4]=0; gz[5]=0; gz[6]=0; gz[7]=0;
            __builtin_amdgcn_tensor_load_to_lds(g0, g1, g2, g3, gz, 0);
#else
            __builtin_amdgcn_tensor_load_to_lds(g0, g1, g2, g3, 0);
#endif
            __builtin_amdgcn_s_wait_tensorcnt(0);
        }
        __syncthreads();                       // all waves reach this barrier
    }

    if (nt * 16 >= N) return;                  // wave-uniform: EXEC stays all-ones
    const int col0 = nt * 16;

    v8f acc;
    #pragma unroll
    for (int j = 0; j < 8; ++j) acc[j] = 0.0f;

    const int    KK    = TDM ? 128 : K;        // compile-time for TDM path
    const size_t stepB = (size_t)4 * N;
    const float* bp    = B + (size_t)ko * N + col0 + m;

    if (TDM) {
        const float* ap = &sA[m * 130 + ko];
        #pragma unroll
        for (int k = 0; k < 128; k += 4) {
            v2f a, b;
            a[0] = ap[k];
            a[1] = ap[k + 1];
            b[0] = bp[0];
            b[1] = bp[N];
            bp += stepB;
            acc = __builtin_amdgcn_wmma_f32_16x16x4_f32(false, a, false, b,
                                                        (short)0, acc, false, false);
        }
    } else {
        const float* ap = A + (size_t)(row0 + m) * KK + ko;
        #pragma unroll 8
        for (int k = 0; k < KK; k += 4) {
            v2f a, b;
            a[0] = ap[k];
            a[1] = ap[k + 1];
            b[0] = bp[0];
            b[1] = bp[N];
            bp += stepB;
            acc = __builtin_amdgcn_wmma_f32_16x16x4_f32(false, a, false, b,
                                                        (short)0, acc, false, false);
        }
    }

    const int rbase = row0 + ((lane >> 4) << 3);   // M = j (+8 for upper half-wave)
    const int col   = col0 + m;
    const float bv  = bias ? bias[col] : 0.0f;
    #pragma unroll
    for (int j = 0; j < 8; ++j) {
        float v = acc[j] + bv;
        if (act == 1) v = siluf_(v);
        C[(size_t)(rbase + j) * N + col] = v;
    }
}

// ---------------------------------------------------------------------------
// Input: h = gelu(LN(x @ Win + b)) + pos_enc[s]       (K=23, per-row block)
// ---------------------------------------------------------------------------
__global__ __launch_bounds__(128) void input_proj(
    const float* __restrict__ x, const float* __restrict__ W,
    const float* __restrict__ bias, const float* __restrict__ g,
    const float* __restrict__ beta, const float* __restrict__ pe,
    float* __restrict__ h)
{
    const int row = blockIdx.x, t = threadIdx.x;
    __shared__ float xs[23];
    __shared__ float red[128];
    if (t < 23) xs[t] = x[row * 23 + t];
    __syncthreads();
    float v = bias[t];
    #pragma unroll
    for (int f = 0; f < 23; ++f) v += xs[f] * W[f * DD + t];
    float mean = block_reduce128(v, red, t) * (1.0f / DD);
    float d    = v - mean;
    float var  = block_reduce128(d * d, red, t) * (1.0f / DD);
    float ln   = d * rsqrtf(var + EPS) * g[t] + beta[t];
    const int s = (row >> 2) & 63;
    h[(size_t)row * DD + t] = geluf_(ln) + pe[s * DD + t];
}

// ---------------------------------------------------------------------------
// eta/al scalar head: out = mul*sigmoid(X[row,:32].W2 + b2) + add (wave/row)
// ---------------------------------------------------------------------------
__global__ __launch_bounds__(256) void head_scalar(
    const float* __restrict__ X, const float* __restrict__ W2,
    const float* __restrict__ b2, float* __restrict__ out,
    float mul, float add)
{
    const int wave = threadIdx.x >> 5, lane = threadIdx.x & 31;
    const int row  = blockIdx.x * 8 + wave;
    float v = X[(size_t)row * 32 + lane] * W2[lane];
    #pragma unroll
    for (int s = 16; s > 0; s >>= 1) v += __shfl_down(v, s, 32);
    if (lane == 0) out[row] = mul * sigmoidf_(v + b2[0]) + add;
}

// ---------------------------------------------------------------------------
// Delta-rule scan: 256 blocks (one per b*n chunk), 128 threads (one per M-row)
// State row M[r,:] lives entirely in VGPRs. ||k|| via wave32 shuffle + 4-way
// LDS combine (2 barriers/step). Next-step operands prefetched.
// ---------------------------------------------------------------------------
__global__ __launch_bounds__(128) void delta_scan(
    const float* __restrict__ Q, const float* __restrict__ Kk,
    const float* __restrict__ VH, const float* __restrict__ ETA,
    const float* __restrict__ AL, float* __restrict__ OUT)
{
    const int i = blockIdx.x;      // chunk
    const int r = threadIdx.x;     // state-matrix row
    __shared__ float sq[128], sk[128], part[4];
    float M[128];
    #pragma unroll
    for (int c = 0; c < 128; ++c) M[c] = 0.0f;

    #pragma unroll 1
    for (int j = 0; j < 64; ++j) {
        const size_t row = (size_t)i * 64 + j;
        const float kk = Kk[row * DD + r];
        sq[r] = Q[row * DD + r];
        sk[r] = kk;
        float v = kk * kk;                           // wave32 reduction of ||k||^2
        #pragma unroll
        for (int s = 16; s > 0; s >>= 1) v += __shfl_xor(v, s, 32);
        if ((r & 31) == 0) part[r >> 5] = v;
        __syncthreads();                             // also fences sq/sk
        const float ks  = part[0] + part[1] + part[2] + part[3];
        const float inv = 1.0f / fmaxf(sqrtf(ks), 1e-12f);

        if (j + 1 < 64) {                            // overlap next-step loads
            __builtin_prefetch(&Q [(row + 1) * DD + r], 0, 3);
            __builtin_prefetch(&Kk[(row + 1) * DD + r], 0, 3);
            __builtin_prefetch(&VH[(row + 1) * DD + r], 0, 3);
        }

        float o = 0.0f, mk = 0.0f;
        #pragma unroll
        for (int c = 0; c < 128; ++c) { o += M[c] * sq[c]; mk += M[c] * sk[c]; }
        OUT[row * DD + r] = o;                       // pre-update M @ q

        const float e = ETA[row], a = AL[row];
        const float coef = e * (VH[row * DD + r] - mk * inv) * inv;
        #pragma unroll
        for (int c = 0; c < 128; ++c) M[c] = a * M[c] + coef * sk[c];
        __syncthreads();                             // before next sq/sk overwrite
    }
}

// h = LN(o + h) * g + b   (in place over h)
__global__ __launch_bounds__(128) void resln(
    float* __restrict__ h, const float* __restrict__ o,
    const float* __restrict__ g, const float* __restrict__ b)
{
    const int row = blockIdx.x, t = threadIdx.x;
    __shared__ float red[128];
    float v    = o[(size_t)row * DD + t] + h[(size_t)row * DD + t];
    float mean = block_reduce128(v, red, t) * (1.0f / DD);
    float d    = v - mean;
    float var  = block_reduce128(d * d, red, t) * (1.0f / DD);
    h[(size_t)row * DD + t] = d * rsqrtf(var + EPS) * g[t] + b[t];
}

// dst = LN(src) * g + b
__global__ __launch_bounds__(128) void ln_write(
    float* __restrict__ dst, const float* __restrict__ src,
    const float* __restrict__ g, const float* __restrict__ b)
{
    const int row = blockIdx.x, t = threadIdx.x;
    __shared__ float red[128];
    float v    = src[(size_t)row * DD + t];
    float mean = block_reduce128(v, red, t) * (1.0f / DD);
    float d    = v - mean;
    float var  = block_reduce128(d * d, red, t) * (1.0f / DD);
    dst[(size_t)row * DD + t] = d * rsqrtf(var + EPS) * g[t] + b[t];
}

// agg (+)= softmax(lw)[idx] * LN(o + h)_level
__global__ __launch_bounds__(128) void cms_acc(
    float* __restrict__ agg, const float* __restrict__ o,
    const float* __restrict__ h, const float* __restrict__ g,
    const float* __restrict__ b, const float* __restrict__ lw, int idx)
{
    const int row = blockIdx.x, t = threadIdx.x;
    __shared__ float red[128];
    float l0 = lw[0], l1 = lw[1], l2 = lw[2], l3 = lw[3];
    float mx = fmaxf(fmaxf(l0, l1), fmaxf(l2, l3));
    float e0 = __expf(l0 - mx), e1 = __expf(l1 - mx), e2 = __expf(l2 - mx), e3 = __expf(l3 - mx);
    float wsel = (idx == 0 ? e0 : idx == 1 ? e1 : idx == 2 ? e2 : e3) / (e0 + e1 + e2 + e3);

    float v    = o[(size_t)row * DD + t] + h[(size_t)row * DD + t];
    float mean = block_reduce128(v, red, t) * (1.0f / DD);
    float d    = v - mean;
    float var  = block_reduce128(d * d, red, t) * (1.0f / DD);
    float ln   = d * rsqrtf(var + EPS) * g[t] + b[t];
    float prev = (idx == 0) ? 0.0f : agg[(size_t)row * DD + t];
    agg[(size_t)row * DD + t] = prev + wsel * ln;
}

// ---------------------------------------------------------------------------
// Regime module (tiny: 64 batches x 4 slots)
// ---------------------------------------------------------------------------
__global__ __launch_bounds__(128) void reg_state(
    const float* __restrict__ h, float* __restrict__ state,
    float* __restrict__ gmean, float* __restrict__ gstd)
{
    const int b = blockIdx.x, t = threadIdx.x;
    float st[4];
    #pragma unroll
    for (int n = 0; n < 4; ++n) {
        float s = 0.0f;
        #pragma unroll
        for (int ss = 61; ss < 64; ++ss)
            s += h[(size_t)(((b * 64) + ss) * 4 + n) * DD + t];
        st[n] = s * (1.0f / 3.0f);
        state[(size_t)(b * 4 + n) * DD + t] = st[n];
    }
    float m = 0.25f * (st[0] + st[1] + st[2] + st[3]);
    float var = 0.0f;
    #pragma unroll
    for (int n = 0; n < 4; ++n) { float d = st[n] - m; var += d * d; }
    var *= (1.0f / 3.0f);                       // ddof = 1
    gmean[b * DD + t] = m;
    gstd[b * DD + t]  = sqrtf(var);
}

__global__ __launch_bounds__(128) void reg_probs(
    const float* __restrict__ state, const float* __restrict__ gmean,
    const float* __restrict__ W1, const float* __restrict__ b1,
    const float* __restrict__ W2, const float* __restrict__ b2,
    float* __restrict__ probs)
{
    const int row = blockIdx.x, t = threadIdx.x, b = row >> 2;
    __shared__ float u[128];
    __shared__ float lg[3];
    float v = b1[t];
    for (int c = 0; c < 128; ++c) v += state[(size_t)row * DD + c] * W1[c * DD + t];
    for (int c = 0; c < 128; ++c) v += gmean[b * DD + c] * W1[(128 + c) * DD + t];
    u[t] = siluf_(v);
    __syncthreads();
    if (t < 3) {
        float z = b2[t];
        for (int c = 0; c < 128; ++c) z += u[c] * W2[c * 3 + t];
        lg[t] = z;
    }
    __syncthreads();
    if (t == 0) {
        float m = fmaxf(lg[0], fmaxf(lg[1], lg[2]));
        float e0 = __expf(lg[0] - m), e1 = __expf(lg[1] - m), e2 = __expf(lg[2] - m);
        float s = e0 + e1 + e2;
        probs[row * 3 + 0] = e0 / s;
        probs[row * 3 + 1] = e1 / s;
        probs[row * 3 + 2] = e2 / s;
    }
}

__global__ __launch_bounds__(128) void reg_alpha(
    const float* __restrict__ state, const float* __restrict__ gmean,
    const float* __restrict__ gstd, const float* __restrict__ probs,
    const float* __restrict__ GG, const float* __restrict__ GB,
    const float* __restrict__ W1, const float* __restrict__ b1,
    const float* __restrict__ W2, const float* __restrict__ b2,
    float* __restrict__ alpha)
{
    const int row = blockIdx.x, t = threadIdx.x, b = row >> 2;
    __shared__ float gi[387];
    __shared__ float red[128];
    __shared__ float z[64];
    gi[t]       = state[(size_t)row * DD + t];
    gi[128 + t] = gmean[b * DD + t];
    gi[256 + t] = gstd[b * DD + t];
    if (t < 3) gi[384 + t] = probs[row * 3 + t];
    __syncthreads();
    float s = 0.0f;
    for (int c = t; c < 387; c += 128) s += gi[c];
    float mean = block_reduce128(s, red, t) * (1.0f / 387.0f);
    float s2 = 0.0f;
    for (int c = t; c < 387; c += 128) { float d = gi[c] - mean; s2 += d * d; }
    float var = block_reduce128(s2, red, t) * (1.0f / 387.0f);
    float inv = rsqrtf(var + EPS);
    for (int c = t; c < 387; c += 128) gi[c] = (gi[c] - mean) * inv * GG[c] + GB[c];
    __syncthreads();
    if (t < 64) {
        float v = b1[t];
        for (int c = 0; c < 387; ++c) v += gi[c] * W1[c * 64 + t];
        z[t] = fmaxf(v, 0.0f);
    }
    __syncthreads();
    if (t == 0) {
        float v = b2[0];
        for (int c = 0; c < 64; ++c) v += z[c] * W2[c];
        alpha[row] = sigmoidf_(v);
    }
}

__global__ __launch_bounds__(128) void reg_attn(
    const float* __restrict__ state, const float* __restrict__ Q,
    const float* __restrict__ K, const float* __restrict__ V,
    const float* __restrict__ alpha, const float* __restrict__ ng,
    const float* __restrict__ nb, float* __restrict__ gout)
{
    const int b = blockIdx.x, t = threadIdx.x;
    __shared__ float sq[4][128], sk[4][128], sv[4][128];
    __shared__ float attn[4][4];
    __shared__ float red[128];
    #pragma unroll
    for (int n = 0; n < 4; ++n) {
        sq[n][t] = Q[(size_t)(b * 4 + n) * DD + t];
        sk[n][t] = K[(size_t)(b * 4 + n) * DD + t];
        sv[n][t] = V[(size_t)(b * 4 + n) * DD + t];
    }
    __syncthreads();
    if (t < 16) {
        int i = t >> 2, j = t & 3;
        float s = 0.0f;
        for (int c = 0; c < 128; ++c) s += sq[i][c] * sk[j][c];
        attn[i][j] = s * 0.08838834764831845f;   // 1/sqrt(128)
    }
    __syncthreads();
    if (t < 4) {
        float m = fmaxf(fmaxf(attn[t][0], attn[t][1]), fmaxf(attn[t][2], attn[t][3]));
        float e[4]; float s = 0.0f;
        #pragma unroll
        for (int j = 0; j < 4; ++j) { e[j] = __expf(attn[t][j] - m); s += e[j]; }
        #pragma unroll
        for (int j = 0; j < 4; ++j) attn[t][j] = e[j] / s;
    }
    __syncthreads();
    for (int i = 0; i < 4; ++i) {
        float a = alpha[b * 4 + i];
        float v = 0.0f;
        #pragma unroll
        for (int j = 0; j < 4; ++j) {
            float mij = a * ((i == j) ? 1.0f : 0.0f) + (1.0f - a) * attn[i][j];
            v += mij * sv[j][t];
        }
        v += state[(size_t)(b * 4 + i) * DD + t];
        float mean = block_reduce128(v, red, t) * (1.0f / DD);
        float d    = v - mean;
        float var  = block_reduce128(d * d, red, t) * (1.0f / DD);
        gout[(size_t)(b * 4 + i) * DD + t] = d * rsqrtf(var + EPS) * ng[t] + nb[t];
        __syncthreads();
    }
}

struct HeadP {
    const float* w1[6]; const float* b1[6];
    const float* w2[6]; const float* b2[6];
};

__global__ __launch_bounds__(128) void final_heads(
    const float* __restrict__ h, const float* __restrict__ greg,
    const float* __restrict__ Wt, const float* __restrict__ bt,
    HeadP hp, float* __restrict__ out)
{
    const int row = blockIdx.x, t = threadIdx.x;
    const int b = row >> 2, n = row & 3;
    __shared__ float comb[128], tt[64], z[32];
    comb[t] = h[(size_t)(((b * 64) + 63) * 4 + n) * DD + t] + greg[(size_t)row * DD + t];
    __syncthreads();
    if (t < 64) {
        float v = bt[t];
        for (int c = 0; c < 128; ++c) v += comb[c] * Wt[c * 64 + t];
        tt[t] = geluf_(v);
    }
    __syncthreads();
    for (int hd = 0; hd < 6; ++hd) {
        if (t < 32) {
            float v = hp.b1[hd][t];
            for (int c = 0; c < 64; ++c) v += tt[c] * hp.w1[hd][c * 32 + t];
            z[t] = geluf_(v);
        }
        __syncthreads();
        if (t == 0) {
            float v = hp.b2[hd][0];
            for (int c = 0; c < 32; ++c) v += z[c] * hp.w2[hd][c];
            float r;
            switch (hd) {
                case 0:  r = tanhf(v); break;                         // direction
                case 1:  r = sigmoidf_(v); break;                     // gate
                case 2:  r = sigmoidf_(v); break;                     // size
                case 3:  r = 0.5f + 5.5f * sigmoidf_(v); break;       // tp
                case 4:  r = 0.3f + 2.7f * sigmoidf_(v); break;       // sl
                default: r = 2.0f + 22.0f * sigmoidf_(v); break;      // hold
            }
            out[hd * 256 + row] = r;
        }
        __syncthreads();
    }
}

// ---------------------------------------------------------------------------
// Host side
// ---------------------------------------------------------------------------
struct Params {
    const float *x;
    const float *inW, *inB, *inG, *inBeta, *pos;
    struct Layer {
        const float *Wq, *Wk, *Wv, *vg1W, *vg1b, *vg2W, *vg2b;
        const float *e1W, *e1b, *e2W, *e2b, *a1W, *a1b, *a2W, *a2b;
        const float *outW, *outb, *g, *b;
    } L[3];
    struct Lvl { const float *l1W, *l1b, *l2W, *l2b, *g, *b; } C[4];
    const float *lw, *fg, *fb;
    const float *rd1W, *rd1b, *rd2W, *rd2b, *qW, *qb, *kW, *kb, *vW, *vb;
    const float *glnG, *glnB, *g1W, *g1b, *g2W, *g2b, *ng, *nb;
    const float *tW, *tb;
    const float *hW1[6], *hB1[6], *hW2[6], *hB2[6];
};

static void fill_params(Params& P, void* const* d, const int* sz) {
    auto F = [&](int i) { return (const float*)d[i]; };
    // insertion-order traversal -> first leaf is x (376832 elems)
    // sorted-key pytree order  -> first leaf is cms.fb (128 elems)
    const bool sorted = sz[0] < 1000;
    if (!sorted) {
        P.x = F(0); P.inW = F(1); P.inB = F(2); P.inG = F(3); P.inBeta = F(4); P.pos = F(5);
        for (int l = 0; l < 3; ++l) { int L = 6 + l * 19; auto& y = P.L[l];
            y.Wq = F(L+0);  y.Wk = F(L+1);  y.Wv = F(L+2);
            y.vg1W = F(L+3); y.vg1b = F(L+4); y.vg2W = F(L+5); y.vg2b = F(L+6);
            y.e1W = F(L+7);  y.e1b = F(L+8);  y.e2W = F(L+9);  y.e2b = F(L+10);
            y.a1W = F(L+11); y.a1b = F(L+12); y.a2W = F(L+13); y.a2b = F(L+14);
            y.outW = F(L+15); y.outb = F(L+16); y.g = F(L+17); y.b = F(L+18); }
        for (int i = 0; i < 4; ++i) { int C = 63 + i * 6; auto& c = P.C[i];
            c.l1W = F(C+0); c.l1b = F(C+1); c.l2W = F(C+2); c.l2b = F(C+3);
            c.g = F(C+4); c.b = F(C+5); }
        P.lw = F(87); P.fg = F(88); P.fb = F(89);
        P.rd1W = F(90); P.rd1b = F(91); P.rd2W = F(92); P.rd2b = F(93);
        P.qW = F(94); P.qb = F(95); P.kW = F(96); P.kb = F(97); P.vW = F(98); P.vb = F(99);
        P.glnG = F(100); P.glnB = F(101); P.g1W = F(102); P.g1b = F(103);
        P.g2W = F(104); P.g2b = F(105); P.ng = F(106); P.nb = F(107);
        P.tW = F(108); P.tb = F(109);
        for (int hd = 0; hd < 6; ++hd) { int H0 = 110 + hd * 4;
            P.hW1[hd] = F(H0); P.hB1[hd] = F(H0+1); P.hW2[hd] = F(H0+2); P.hB2[hd] = F(H0+3); }
    } else {
        P.fb = F(0); P.fg = F(1);
        for (int i = 0; i < 4; ++i) { int C = 2 + i * 6; auto& c = P.C[i];
            c.b = F(C+0); c.g = F(C+1); c.l1W = F(C+2); c.l1b = F(C+3);
            c.l2W = F(C+4); c.l2b = F(C+5); }
        P.lw = F(26);
        // sorted head order: dir(27) gate(31) hold(35) size(39) sl(43) tp(47)
        const int hb[6] = {27, 31, 39, 47, 43, 35};  // slots: dir,gate,size,tp,sl,hold
        for (int hd = 0; hd < 6; ++hd) { int H0 = hb[hd];
            P.hW1[hd] = F(H0); P.hB1[hd] = F(H0+1); P.hW2[hd] = F(H0+2); P.hB2[hd] = F(H0+3); }
        P.inW = F(51); P.inB = F(52); P.inBeta = F(53); P.inG = F(54);
        for (int l = 0; l < 3; ++l) { int L = 55 + l * 19; auto& y = P.L[l];
            y.Wk = F(L+0); y.Wq = F(L+1); y.Wv = F(L+2);
            y.a1W = F(L+3); y.a1b = F(L+4); y.a2W = F(L+5); y.a2b = F(L+6);
            y.b = F(L+7);
            y.e1W = F(L+8); y.e1b = F(L+9); y.e2W = F(L+10); y.e2b = F(L+11);
            y.g = F(L+12); y.outW = F(L+13); y.outb = F(L+14);
            y.vg1W = F(L+15); y.vg1b = F(L+16); y.vg2W = F(L+17); y.vg2b = F(L+18); }
        P.pos = F(112);
        P.g1W = F(113); P.g1b = F(114); P.g2W = F(115); P.g2b = F(116);
        P.glnB = F(117); P.glnG = F(118); P.kW = F(119); P.kb = F(120);
        P.nb = F(121); P.ng = F(122); P.qW = F(123); P.qb = F(124);
        P.rd1W = F(125); P.rd1b = F(126); P.rd2W = F(127); P.rd2b = F(128);
        P.vW = F(129); P.vb = F(130); P.tW = F(131); P.tb = F(132); P.x = F(133);
    }
}

static inline void launch_gemm(const float* A, const float* B, const float* bias,
                               float* C, int M, int N, int K, int act, hipStream_t s) {
    int ctiles = N / 16;
    int wpb = ctiles < 8 ? ctiles : 8;
    dim3 blk(wpb * 32);
    dim3 grd(M / 16, (ctiles + wpb - 1) / wpb);
    if (K == 128)
        gemm_wmma_f32<true><<<grd, blk, 0, s>>>(A, B, bias, C, M, N, K, act);
    else
        gemm_wmma_f32<false><<<grd, blk, 0, s>>>(A, B, bias, C, M, N, K, act);
}

extern "C" void kernel_launch(void* const* d_in, const int* in_sizes, int n_in,
                              void* d_out, int out_size, void* d_ws, size_t ws_size,
                              hipStream_t stream) {
    (void)n_in; (void)out_size; (void)ws_size;
    Params P;
    fill_params(P, d_in, in_sizes);

    const size_t R = ROWS;
    float* W   = (float*)d_ws;
    float* H   = W;                       // R*128  activations
    float* Qb  = H   + R * 128;           // R*128
    float* Kb  = Qb  + R * 128;           // R*128
    float* Vb  = Kb  + R * 128;           // R*128
    float* Ob  = Vb  + R * 128;           // R*128
    float* AGG = Ob  + R * 128;           // R*128
    float* Tb  = AGG + R * 128;           // R*512  big temp
    float* E1  = Tb  + R * 512;           // R*32
    float* A1  = E1  + R * 32;            // R*32
    float* ETA = A1  + R * 32;            // R
    float* AL  = ETA + R;                 // R
    float* Sst = AL  + R;                 // 256*128
    float* Gm  = Sst + 256 * 128;         // 64*128
    float* Gs  = Gm  + 64 * 128;          // 64*128
    float* Pr  = Gs  + 64 * 128;          // 256*3
    float* Alp = Pr  + 768;               // 256
    float* Qr  = Alp + 256;               // 256*128
    float* Kr  = Qr  + 256 * 128;         // 256*128
    float* Vr  = Kr  + 256 * 128;         // 256*128
    float* Gg  = Vr  + 256 * 128;         // 256*128

    // ---- input projection + LN + gelu + pos_enc ----
    input_proj<<<ROWS, 128, 0, stream>>>(P.x, P.inW, P.inB, P.inG, P.inBeta, P.pos, H);

    // ---- 3 delta-memory layers ----
    for (int l = 0; l < 3; ++l) {
        const Params::Layer& y = P.L[l];
        launch_gemm(H,  y.Wq,   nullptr, Qb, ROWS, 128, 128, 0, stream);
        launch_gemm(H,  y.Wk,   nullptr, Kb, ROWS, 128, 128, 0, stream);
        launch_gemm(H,  y.Wv,   nullptr, Vb, ROWS, 128, 128, 0, stream);
        launch_gemm(Vb, y.vg1W, y.vg1b,  Tb, ROWS, 128, 128, 1, stream);  // silu
        launch_gemm(Tb, y.vg2W, y.vg2b,  Vb, ROWS, 128, 128, 0, stream);  // vh
        launch_gemm(H,  y.e1W,  y.e1b,   E1, ROWS, 32, 128, 1, stream);
        launch_gemm(H,  y.a1W,  y.a1b,   A1, ROWS, 32, 128, 1, stream);
        head_scalar<<<ROWS / 8, 256, 0, stream>>>(E1, y.e2W, y.e2b, ETA, 0.1f, 0.01f);
        head_scalar<<<ROWS / 8, 256, 0, stream>>>(A1, y.a2W, y.a2b, AL, 0.5f, 0.5f);
        delta_scan<<<256, 128, 0, stream>>>(Qb, Kb, Vb, ETA, AL, Ob);
        launch_gemm(Ob, y.outW, y.outb, Tb, ROWS, 128, 128, 0, stream);
        resln<<<ROWS, 128, 0, stream>>>(H, Tb, y.g, y.b);
    }

    // ---- continuum memory MLP (4 levels) ----
    for (int i = 0; i < 4; ++i) {
        const Params::Lvl& c = P.C[i];
        launch_gemm(H,  c.l1W, c.l1b, Tb, ROWS, 512, 128, 1, stream);     // silu
        launch_gemm(Tb, c.l2W, c.l2b, Ob, ROWS, 128, 512, 0, stream);
        cms_acc<<<ROWS, 128, 0, stream>>>(AGG, Ob, H, c.g, c.b, P.lw, i);
    }
    ln_write<<<ROWS, 128, 0, stream>>>(H, AGG, P.fg, P.fb);

    // ---- regime module ----
    reg_state<<<64, 128, 0, stream>>>(H, Sst, Gm, Gs);
    reg_probs<<<256, 128, 0, stream>>>(Sst, Gm, P.rd1W, P.rd1b, P.rd2W, P.rd2b, Pr);
    reg_alpha<<<256, 128, 0, stream>>>(Sst, Gm, Gs, Pr, P.glnG, P.glnB,
                                       P.g1W, P.g1b, P.g2W, P.g2b, Alp);
    launch_gemm(Sst, P.qW, P.qb, Qr, 256, 128, 128, 0, stream);
    launch_gemm(Sst, P.kW, P.kb, Kr, 256, 128, 128, 0, stream);
    launch_gemm(Sst, P.vW, P.vb, Vr, 256, 128, 128, 0, stream);
    reg_attn<<<64, 128, 0, stream>>>(Sst, Qr, Kr, Vr, Alp, P.ng, P.nb, Gg);

    // ---- trunk + heads ----
    HeadP hp;
    for (int hd = 0; hd < 6; ++hd) {
        hp.w1[hd] = P.hW1[hd]; hp.b1[hd] = P.hB1[hd];
        hp.w2[hd] = P.hW2[hd]; hp.b2[hd] = P.hB2[hd];
    }
    final_heads<<<256, 128, 0, stream>>>(H, Gg, P.tW, P.tb, hp, (float*)d_out);
}